// BahdanauAttention_78271484003129
// MI455X (gfx1250) — compile-verified
//
#include <hip/hip_runtime.h>
#include <hip/hip_bf16.h>

typedef __attribute__((ext_vector_type(2))) float v2f;
typedef __attribute__((ext_vector_type(4))) float v4f;
typedef __attribute__((ext_vector_type(8))) float v8f;

#define SEQ 2048
#define BAT 16
#define HID 1024

// ---------------------------------------------------------------------------
// Kernel 1: energy[b,n] = sum_k state[b,k] * W[n,k] + bias[n]
// M=16 (batch), N=1024 (tiled by 16), K=1024 (stepped by 4) via
// V_WMMA_F32_16X16X4_F32. One wave per 16-wide N tile -> 64 waves total.
// ---------------------------------------------------------------------------
__global__ __launch_bounds__(32) void energy_wmma_kernel(
    const float* __restrict__ state,  // [16, 1024]  (= last_decoder_state[0,0])
    const float* __restrict__ W,      // [1024, 1024] row-major (n, k)
    const float* __restrict__ bias,   // [1024]
    float* __restrict__ energy)       // [16, 1024]
{
    const int lane  = threadIdx.x;        // 0..31, EXEC all ones
    const int half  = lane >> 4;          // 0 | 1
    const int l16   = lane & 15;
    const int nBase = blockIdx.x * 16;

    // A 16x4 f32 frag: lane<16 holds M=l16, K={k,k+1}; lane>=16 holds M=l16, K={k+2,k+3}
    const float* aPtr = state + l16 * HID + 2 * half;
    // B 4x16 f32 frag (B[k][n] = W[n][k]): mirrored layout, lane gives N=nBase+l16
    const float* bPtr = W + (size_t)(nBase + l16) * HID + 2 * half;

    v8f acc = {};
    for (int k = 0; k < HID; k += 4) {
        v2f a, b;
        a.x = aPtr[k];     a.y = aPtr[k + 1];
        b.x = bPtr[k];     b.y = bPtr[k + 1];
        acc = __builtin_amdgcn_wmma_f32_16x16x4_f32(
            /*neg_a=*/false, a, /*neg_b=*/false, b,
            /*c_mod=*/(short)0, acc, /*reuse_a=*/false, /*reuse_b=*/false);
    }

    // D 16x16 f32 layout: VGPR r -> M = r + 8*half, N = nBase + l16
    const int n  = nBase + l16;
    const float bn = bias[n];
#pragma unroll
    for (int r = 0; r < 8; ++r) {
        const int m = r + 8 * half;
        energy[m * HID + n] = acc[r] + bn;
    }
}

// ---------------------------------------------------------------------------
// Kernel 2: scores[s,b] = dot(enc[s,b,:], energy[b,:]).  Streams 128 MB once
// -> non-temporal b128 loads. One wave per (s,b); 8 waves/block; block covers
// 8 consecutive b for one s; two blocks per s.
// ---------------------------------------------------------------------------
__global__ __launch_bounds__(256) void scores_kernel(
    const float* __restrict__ enc,     // [S, B, H]
    const float* __restrict__ energy,  // [B, H]
    float* __restrict__ scores)        // [S, B]
{
    const int wave = threadIdx.x >> 5;
    const int lane = threadIdx.x & 31;
    const int s    = blockIdx.x >> 1;
    const int b    = ((blockIdx.x & 1) << 3) + wave;

    const float* e = enc + ((size_t)s * BAT + b) * HID;
    const float* g = energy + b * HID;

    float acc = 0.f;
#pragma unroll
    for (int i = 0; i < 8; ++i) {
        const int h = i * 128 + lane * 4;                 // wave covers 512B/step
        v4f ev = __builtin_nontemporal_load((const v4f*)(e + h));  // stream, don't cache
        v4f gv = *(const v4f*)(g + h);                    // hot, L2-resident
        acc += ev.x * gv.x + ev.y * gv.y + ev.z * gv.z + ev.w * gv.w;
    }
#pragma unroll
    for (int off = 16; off > 0; off >>= 1)
        acc += __shfl_down(acc, off, 32);

    if (lane == 0) scores[s * BAT + b] = acc;
}

// ---------------------------------------------------------------------------
// Kernel 3: out[s,b] = softmax over s of scores[s,b].  One block per column b.
// ---------------------------------------------------------------------------
__global__ __launch_bounds__(256) void softmax_kernel(
    const float* __restrict__ scores,  // [S, B]
    float* __restrict__ out)           // [1,1,S,B] flat = [S, B]
{
    const int b = blockIdx.x;
    const int t = threadIdx.x;
    __shared__ float red[256];

    float v[8];
    float m = -INFINITY;
#pragma unroll
    for (int i = 0; i < 8; ++i) {
        v[i] = scores[(i * 256 + t) * BAT + b];
        m = fmaxf(m, v[i]);
    }
    red[t] = m;
    __syncthreads();
    for (int off = 128; off > 0; off >>= 1) {
        if (t < off) red[t] = fmaxf(red[t], red[t + off]);
        __syncthreads();
    }
    m = red[0];
    __syncthreads();

    float sum = 0.f;
#pragma unroll
    for (int i = 0; i < 8; ++i) {
        v[i] = expf(v[i] - m);
        sum += v[i];
    }
    red[t] = sum;
    __syncthreads();
    for (int off = 128; off > 0; off >>= 1) {
        if (t < off) red[t] += red[t + off];
        __syncthreads();
    }
    const float inv = 1.0f / red[0];
#pragma unroll
    for (int i = 0; i < 8; ++i)
        out[(i * 256 + t) * BAT + b] = v[i] * inv;
}

// ---------------------------------------------------------------------------
extern "C" void kernel_launch(void* const* d_in, const int* in_sizes, int n_in,
                              void* d_out, int out_size, void* d_ws, size_t ws_size,
                              hipStream_t stream) {
    (void)in_sizes; (void)n_in; (void)out_size; (void)ws_size;

    const float* enc   = (const float*)d_in[0];   // [2048, 16, 1024]
    const float* state = (const float*)d_in[1];   // [2, 1, 16, 1024]; [0,0] = first 16*1024
    const float* W     = (const float*)d_in[2];   // [1024, 1024]
    const float* bias  = (const float*)d_in[3];   // [1024]
    float*       out   = (float*)d_out;           // 2048*16

    float* energy = (float*)d_ws;                 // 16*1024 floats  (64 KB)
    float* scores = energy + BAT * HID;           // 2048*16 floats  (128 KB)

    energy_wmma_kernel<<<HID / 16, 32, 0, stream>>>(state, W, bias, energy);
    scores_kernel<<<SEQ * 2, 256, 0, stream>>>(enc, energy, scores);
    softmax_kernel<<<BAT, 256, 0, stream>>>(scores, out);
}